// GraphModel_50225347559473
// MI455X (gfx1250) — compile-verified
//
#include <hip/hip_runtime.h>

typedef __attribute__((ext_vector_type(2))) float v2f;
typedef __attribute__((ext_vector_type(8))) float v8f;

#define M_ROWS 10000

// ---------------------------------------------------------------------------
// Pass 1: per-row scale factor.
// scale[i] = prod_j (1 + w0*e_i + w1*e_j + b) / (1 + (w0+w1)*e_i + b)
// O(N^2) VALU pass; e[] tiles staged through LDS (broadcast reuse).
// ---------------------------------------------------------------------------
__global__ __launch_bounds__(256)
void scale_kernel(const float* __restrict__ e,
                  const float* __restrict__ Ew,
                  const float* __restrict__ Eb,
                  float* __restrict__ scale, int n)
{
    __shared__ float tj[256];
    const int i = blockIdx.x * 256 + threadIdx.x;
    const float w0 = Ew[0], w1 = Ew[1], b = Eb[0];
    const float ei = (i < n) ? e[i] : 0.0f;
    const float a = 1.0f + w0 * ei + b;     // row-constant part of (1 + Edge)
    float prod = 1.0f;
    for (int jt = 0; jt < n; jt += 256) {
        const int j = jt + threadIdx.x;
        tj[threadIdx.x] = (j < n) ? (w1 * e[j]) : 0.0f;
        __syncthreads();
        const int lim = min(256, n - jt);
        for (int k = 0; k < lim; ++k)
            prod *= (a + tj[k]);
        __syncthreads();
    }
    if (i < n) {
        const float diag = 1.0f + (w0 + w1) * ei + b;
        scale[i] = prod / diag;
    }
}

// ---------------------------------------------------------------------------
// Fused layer: O = relu(A @ W^T + bias) * scale[:,None]
// A: (M,K) row-major, W: (NOUT,K) row-major. One 16x16 output tile per wave,
// K-loop of V_WMMA_F32_16X16X4_F32 (full fp32, matches reference precision).
//
// Fragment layouts (ISA 7.12.2, 32-bit operands, wave32):
//   A 16x4 : lane L -> row M = L&15 ; VGPR0/1 = K = (L>>4)*2 + {0,1}
//   B 4x16 : lane L -> col N = L&15 ; VGPR0/1 = K = (L>>4)*2 + {0,1}
//   C/D    : VGPR v, lane L -> row = v + 8*(L>>4), col = L&15
// ---------------------------------------------------------------------------
template <int K, int NOUT>
__global__ __launch_bounds__(256)
void wmma_layer_kernel(const float* __restrict__ A,
                       const float* __restrict__ W,
                       const float* __restrict__ bias,
                       const float* __restrict__ scale,
                       float* __restrict__ O)
{
    constexpr int TILES_N = NOUT / 16;
    constexpr int TILES_M = M_ROWS / 16;

    const int wave = blockIdx.x * (blockDim.x >> 5) + (threadIdx.x >> 5);
    if (wave >= TILES_M * TILES_N) return;          // wave-uniform: EXEC stays all-ones

    const int tM = wave / TILES_N;
    const int tN = wave - tM * TILES_N;

    const int lane = threadIdx.x & 31;
    const int half = lane >> 4;                      // selects K pair {0,1} vs {2,3}
    const int l15  = lane & 15;

    const float* arow = A + (size_t)(tM * 16 + l15) * K + half * 2;
    const float* wrow = W + (size_t)(tN * 16 + l15) * K + half * 2;

    v8f acc = {};
#pragma unroll
    for (int k = 0; k < K; k += 4) {
        const v2f av = *(const v2f*)(arow + k);      // 8B aligned: K even, k%4==0
        const v2f bv = *(const v2f*)(wrow + k);
        acc = __builtin_amdgcn_wmma_f32_16x16x4_f32(
                  false, av, false, bv, (short)0, acc, false, false);
    }

    const int   col = tN * 16 + l15;
    const float bn  = bias[col];
#pragma unroll
    for (int v = 0; v < 8; ++v) {
        const int row = tM * 16 + v + 8 * half;
        const float r = fmaxf(acc[v] + bn, 0.0f) * scale[row];
        O[(size_t)row * NOUT + col] = r;
    }
}

// ---------------------------------------------------------------------------
// Final layer: NOUT=2, K=32 — trivial per-row dot products (not worth a
// padded WMMA tile).  out[i][n] = relu(h.W4[n] + b4[n]) * scale[i]
// ---------------------------------------------------------------------------
__global__ __launch_bounds__(256)
void last_layer_kernel(const float* __restrict__ h,      // (M,32)
                       const float* __restrict__ W,      // (2,32)
                       const float* __restrict__ bias,   // (2,)
                       const float* __restrict__ scale,
                       float* __restrict__ out, int n)
{
    const int i = blockIdx.x * blockDim.x + threadIdx.x;
    if (i >= n) return;
    const float* hr = h + (size_t)i * 32;
    float acc0 = bias[0], acc1 = bias[1];
#pragma unroll
    for (int k = 0; k < 32; ++k) {
        const float hv = hr[k];
        acc0 = fmaf(hv, W[k], acc0);
        acc1 = fmaf(hv, W[32 + k], acc1);
    }
    const float s = scale[i];
    out[2 * i + 0] = fmaxf(acc0, 0.0f) * s;
    out[2 * i + 1] = fmaxf(acc1, 0.0f) * s;
}

// ---------------------------------------------------------------------------
extern "C" void kernel_launch(void* const* d_in, const int* in_sizes, int n_in,
                              void* d_out, int out_size, void* d_ws, size_t ws_size,
                              hipStream_t stream)
{
    const float* node = (const float*)d_in[0];   // (10000,100)
    const float* edge = (const float*)d_in[1];   // (10000,)
    const float* Ew   = (const float*)d_in[2];   // (2,)
    const float* Eb   = (const float*)d_in[3];   // ()
    const float* W0   = (const float*)d_in[4];   const float* b0 = (const float*)d_in[5];
    const float* W1   = (const float*)d_in[6];   const float* b1 = (const float*)d_in[7];
    const float* W2   = (const float*)d_in[8];   const float* b2 = (const float*)d_in[9];
    const float* W3   = (const float*)d_in[10];  const float* b3 = (const float*)d_in[11];
    const float* W4   = (const float*)d_in[12];  const float* b4 = (const float*)d_in[13];
    float* out = (float*)d_out;

    const int M = M_ROWS;

    // Workspace partition (ping-pong): scale | bufA (M*256) | bufB (M*112)
    char* ws = (char*)d_ws;
    float* scale = (float*)ws;       ws += ((size_t)M * sizeof(float) + 255) & ~(size_t)255;
    float* bufA  = (float*)ws;       ws += (size_t)M * 256 * sizeof(float);
    float* bufB  = (float*)ws;       // M*112 floats (also reused for M*32)

    float* h1 = bufA;  // (M,256)
    float* h2 = bufB;  // (M,112)
    float* h3 = bufA;  // (M, 64)  overwrites h1 (dead)
    float* h4 = bufB;  // (M, 32)  overwrites h2 (dead)

    scale_kernel<<<(M + 255) / 256, 256, 0, stream>>>(edge, Ew, Eb, scale, M);

    {   // layer 0: 100 -> 256
        constexpr int tiles = (M_ROWS / 16) * (256 / 16);
        wmma_layer_kernel<100, 256><<<(tiles + 7) / 8, 256, 0, stream>>>(node, W0, b0, scale, h1);
    }
    {   // layer 1: 256 -> 112
        constexpr int tiles = (M_ROWS / 16) * (112 / 16);
        wmma_layer_kernel<256, 112><<<(tiles + 7) / 8, 256, 0, stream>>>(h1, W1, b1, scale, h2);
    }
    {   // layer 2: 112 -> 64
        constexpr int tiles = (M_ROWS / 16) * (64 / 16);
        wmma_layer_kernel<112, 64><<<(tiles + 7) / 8, 256, 0, stream>>>(h2, W2, b2, scale, h3);
    }
    {   // layer 3: 64 -> 32
        constexpr int tiles = (M_ROWS / 16) * (32 / 16);
        wmma_layer_kernel<64, 32><<<(tiles + 7) / 8, 256, 0, stream>>>(h3, W3, b3, scale, h4);
    }
    // layer 4: 32 -> 2
    last_layer_kernel<<<(M + 255) / 256, 256, 0, stream>>>(h4, W4, b4, scale, out, M);
}